// MLLM_91061896610192
// MI455X (gfx1250) — compile-verified
//
#include <hip/hip_runtime.h>
#include <math.h>

typedef float v2f __attribute__((ext_vector_type(2)));
typedef float v8f __attribute__((ext_vector_type(8)));

#define DIM 100
#define SEQ 128
#define BAT 16
#define NPOS (BAT * SEQ)   // 2048
#define LD 208             // 2*DIM = 200 padded to 13*16
#define MT 13
#define NT (NPOS / 16)     // 128
#define WMAT_SZ (LD * LD)  // 43264

// ---------------------------------------------------------------------------
// Build stacked real GEMM matrices from complex kernels:
//   rows 0..99   : [  Vr | Vi ] / ||v_k||      ("a" part: Re<v,phi> pieces)
//   rows 100..199: [ -Vi | Vr ] / ||v_k||      ("c" part: Im<v,phi> pieces)
//   rows/cols 200..207: zero padding
// ---------------------------------------------------------------------------
__global__ void prep_weights(const float* __restrict__ pr, const float* __restrict__ pi,
                             const float* __restrict__ mr, const float* __restrict__ mim,
                             float* __restrict__ W) {
    int mat = blockIdx.x / LD;   // 0,1 = proj layers, 2 = measurement
    int m   = blockIdx.x % LD;
    int tid = threadIdx.x;       // 64 threads
    float* row = W + (size_t)mat * WMAT_SZ + (size_t)m * LD;
    if (m >= 2 * DIM) {
        for (int c = tid; c < LD; c += 64) row[c] = 0.0f;
        return;
    }
    int k = (m < DIM) ? m : m - DIM;
    const float* vr = (mat < 2) ? pr + (size_t)mat * DIM * DIM + (size_t)k * DIM
                                : mr + (size_t)k * DIM;
    const float* vi = (mat < 2) ? pi + (size_t)mat * DIM * DIM + (size_t)k * DIM
                                : mim + (size_t)k * DIM;
    __shared__ float red[64];
    float ss = 0.0f;
    for (int d = tid; d < DIM; d += 64) ss += vr[d] * vr[d] + vi[d] * vi[d];
    red[tid] = ss;
    __syncthreads();
    for (int off = 32; off > 0; off >>= 1) {
        if (tid < off) red[tid] += red[tid + off];
        __syncthreads();
    }
    float inv = 1.0f / sqrtf(red[0]);
    bool apart = (m < DIM);
    for (int c = tid; c < LD; c += 64) {
        float v;
        if (c < DIM)          v = (apart ? vr[c] : -vi[c]) * inv;
        else if (c < 2 * DIM) v = (apart ? vi[c - DIM] : vr[c - DIM]) * inv;
        else                  v = 0.0f;
        row[c] = v;
    }
}

// ---------------------------------------------------------------------------
// Embedding gather + L2 norm + complex state.  X[pos][0..99]=r, [100..199]=i,
// [200..207]=0.  wbuf[pos] = raw L2 norm (mixture weight).
// ---------------------------------------------------------------------------
__global__ void embed(const int* __restrict__ seq, const float* __restrict__ ampT,
                      const float* __restrict__ phT, float* __restrict__ X,
                      float* __restrict__ wbuf) {
    int pos = blockIdx.x;
    int tid = threadIdx.x;  // 128
    int v = seq[pos];
    float a = 0.0f, ph = 0.0f;
    if (tid < DIM) {
        a  = ampT[(size_t)v * DIM + tid];
        ph = phT[(size_t)v * DIM + tid];
    }
    __shared__ float red[128];
    red[tid] = a * a;
    __syncthreads();
    for (int off = 64; off > 0; off >>= 1) {
        if (tid < off) red[tid] += red[tid + off];
        __syncthreads();
    }
    float w = sqrtf(red[0]);
    if (tid == 0) wbuf[pos] = w;
    float* xr = X + (size_t)pos * LD;
    if (tid < DIM) {
        float an = a / w;
        xr[tid]       = an * cosf(ph);
        xr[DIM + tid] = an * sinf(ph);
    } else if (tid < DIM + 8) {
        xr[DIM + tid] = 0.0f;  // pad cols 200..207
    }
}

// ---------------------------------------------------------------------------
// Per-batch softmax machinery: E[t] = exp(w-t max), den[j] for window offsets
// j=0..2 (softmax over the sequence axis includes j padded exp(0) entries).
// ---------------------------------------------------------------------------
__global__ void softmax_prep(const float* __restrict__ wbuf, float* __restrict__ E,
                             float* __restrict__ den) {
    int b = blockIdx.x;
    int t = threadIdx.x;  // 128
    __shared__ float sh[128];
    __shared__ float e01[2];
    float w = wbuf[b * SEQ + t];
    sh[t] = w;
    __syncthreads();
    for (int off = 64; off > 0; off >>= 1) {
        if (t < off) sh[t] = fmaxf(sh[t], sh[t + off]);
        __syncthreads();
    }
    float mx = sh[0];
    __syncthreads();
    float e = expf(w - mx);
    E[b * SEQ + t] = e;
    if (t < 2) e01[t] = e;
    sh[t] = e;
    __syncthreads();
    for (int off = 64; off > 0; off >>= 1) {
        if (t < off) sh[t] += sh[t + off];
        __syncthreads();
    }
    if (t == 0) {
        float total = sh[0];
        float ez = expf(0.0f - mx);  // padded entries
        den[b * 3 + 0] = total;
        den[b * 3 + 1] = total - e01[0] + ez;
        den[b * 3 + 2] = total - e01[0] - e01[1] + 2.0f * ez;
    }
}

// ---------------------------------------------------------------------------
// GEMM: Out[N=2048][M=208] = X[N][K=208] * W[M][K]^T via V_WMMA_F32_16X16X4_F32.
// One wave per 16x16 tile, K swept in steps of 4 (full f32 precision).
// A lane layout: lane L -> row (L&15), cols k+2*(L>>4).  Same for B (=X rows).
// D lane layout: VGPR v, lane L -> (M = v + 8*(L>>4), N = L&15).
// ---------------------------------------------------------------------------
__global__ void gemm_wmma(const float* __restrict__ W, const float* __restrict__ Xm,
                          float* __restrict__ Out) {
    int wave = threadIdx.x >> 5;
    int lane = threadIdx.x & 31;
    int tile = blockIdx.x * 8 + wave;       // 208 blocks * 8 waves = 1664 tiles
    int mt = tile / NT;
    int nt = tile - mt * NT;
    int m0 = mt * 16, n0 = nt * 16;
    int half = lane >> 4;
    int l = lane & 15;
    const float* Ap = W  + (size_t)(m0 + l) * LD + 2 * half;
    const float* Bp = Xm + (size_t)(n0 + l) * LD + 2 * half;
    v8f acc = {0.f, 0.f, 0.f, 0.f, 0.f, 0.f, 0.f, 0.f};
    for (int k = 0; k < LD; k += 4) {
        v2f av = *(const v2f*)(Ap + k);
        v2f bv = *(const v2f*)(Bp + k);
        acc = __builtin_amdgcn_wmma_f32_16x16x4_f32(false, av, false, bv,
                                                    (short)0, acc, false, false);
    }
    float* o = Out + (size_t)(n0 + l) * LD + m0 + 8 * half;
#pragma unroll
    for (int i = 0; i < 8; ++i) o[i] = acc[i];
}

// ---------------------------------------------------------------------------
// n-gram combine: new_r[pos][k] = sum_j gw(b,s,j) * (a[pos+j][k]^2 + c[pos+j][k]^2)
// (Im<v|rho|v> == 0 exactly since rho is Hermitian), new_i = 0.
// ---------------------------------------------------------------------------
__global__ void combine(const float* __restrict__ Out, const float* __restrict__ E,
                        const float* __restrict__ den, float* __restrict__ X, int n) {
    int pos = blockIdx.x;
    int b = pos >> 7;
    int s = pos & (SEQ - 1);
    int tid = threadIdx.x;  // 128
    float* xr = X + (size_t)pos * LD;
    if (tid < DIM) {
        float acc = 0.0f;
        for (int j = 0; j < n; ++j) {
            if (s + j < SEQ) {  // padded windows have phi = 0 -> contribute 0
                float qa = Out[(size_t)(pos + j) * LD + tid];
                float qc = Out[(size_t)(pos + j) * LD + DIM + tid];
                float gw = E[b * SEQ + s + j] / den[b * 3 + j];
                acc += gw * (qa * qa + qc * qc);
            }
        }
        xr[tid]       = acc;
        xr[DIM + tid] = 0.0f;
    } else if (tid < DIM + 8) {
        xr[DIM + tid] = 0.0f;
    }
}

// ---------------------------------------------------------------------------
// Final mixture over the sequence with raw L2-norm weights, then dense head.
// ---------------------------------------------------------------------------
__global__ void final_reduce(const float* __restrict__ Out, const float* __restrict__ wbuf,
                             const float* __restrict__ dw, const float* __restrict__ db,
                             float* __restrict__ out) {
    int b = blockIdx.x;
    int tid = threadIdx.x;  // 128
    __shared__ float probs[DIM];
    if (tid < DIM) {
        float acc = 0.0f;
        for (int s = 0; s < SEQ; ++s) {
            const float* row = Out + (size_t)(b * SEQ + s) * LD;
            float qa = row[tid], qc = row[DIM + tid];
            acc += wbuf[b * SEQ + s] * (qa * qa + qc * qc);
        }
        probs[tid] = acc;
    }
    __syncthreads();
    if (tid < 2) {
        float o = db[tid];
        for (int k = 0; k < DIM; ++k) o += probs[k] * dw[k * 2 + tid];
        out[b * 2 + tid] = o;
    }
}

extern "C" void kernel_launch(void* const* d_in, const int* in_sizes, int n_in,
                              void* d_out, int out_size, void* d_ws, size_t ws_size,
                              hipStream_t stream) {
    const int*   seq  = (const int*)d_in[0];
    const float* ampT = (const float*)d_in[1];
    const float* phT  = (const float*)d_in[2];
    const float* pr   = (const float*)d_in[3];
    const float* pi   = (const float*)d_in[4];
    const float* mr   = (const float*)d_in[5];
    const float* mim  = (const float*)d_in[6];
    const float* dw   = (const float*)d_in[7];
    const float* db   = (const float*)d_in[8];

    float* ws  = (float*)d_ws;
    float* X   = ws;                                 // 2048*208
    float* Out = ws + (size_t)NPOS * LD;             // 2048*208
    float* W   = ws + (size_t)2 * NPOS * LD;         // 3*208*208
    float* wb  = W + (size_t)3 * WMAT_SZ;            // 2048
    float* E   = wb + NPOS;                          // 2048
    float* den = E + NPOS;                           // 48

    prep_weights<<<3 * LD, 64, 0, stream>>>(pr, pi, mr, mim, W);
    embed<<<NPOS, 128, 0, stream>>>(seq, ampT, phT, X, wb);
    softmax_prep<<<BAT, 128, 0, stream>>>(wb, E, den);

    gemm_wmma<<<(MT * NT) / 8, 256, 0, stream>>>(W, X, Out);               // layer 0 (n=2)
    combine<<<NPOS, 128, 0, stream>>>(Out, E, den, X, 2);
    gemm_wmma<<<(MT * NT) / 8, 256, 0, stream>>>(W + WMAT_SZ, X, Out);     // layer 1 (n=3)
    combine<<<NPOS, 128, 0, stream>>>(Out, E, den, X, 3);
    gemm_wmma<<<(MT * NT) / 8, 256, 0, stream>>>(W + 2 * WMAT_SZ, X, Out); // measurement
    final_reduce<<<BAT, 128, 0, stream>>>(Out, wb, dw, db, (float*)d_out);
}